// GPTNeoXAttention_7344394076211
// MI455X (gfx1250) — compile-verified
//
#include <hip/hip_runtime.h>

typedef __attribute__((ext_vector_type(2))) float v2f;
typedef __attribute__((ext_vector_type(8))) float v8f;

#define B_      16
#define S_      2048
#define NH_     32
#define HD_     128
#define HIDDEN_ 4096
#define QKV_N   (3 * HIDDEN_)
#define ROT_    32
#define HALFROT 16

// ---------------------------------------------------------------------------
// C(16 x N) = A(16 x K) * W(K x N) + bias
// One 16x16 output tile per wave32, accumulated with V_WMMA_F32_16X16X4_F32.
//
// ISA layouts (CDNA5 ISA §7.12.2):
//  A 16x4 f32 : lanes 0-15 hold M=lane; VGPR0/1 = K{0,1}; lanes 16-31 = K{2,3}
//  B 4x16 f32 : lanes 0-15 hold N=lane; VGPR0/1 = K{0,1}; lanes 16-31 = K{2,3}
//  C/D 16x16  : lanes 0-15 N=lane; VGPR r = M=r (lanes 0-15), M=r+8 (lanes 16-31)
// ---------------------------------------------------------------------------
__global__ __launch_bounds__(256)
void gemm16_wmma(const float* __restrict__ A, const float* __restrict__ W,
                 const float* __restrict__ bias, float* __restrict__ C,
                 int K, int N) {
  const int wave = threadIdx.x >> 5;
  const int lane = threadIdx.x & 31;
  const int tile = blockIdx.x * 8 + wave;
  const int col0 = tile * 16;
  if (col0 >= N) return;                 // uniform per wave: EXEC all-1s inside

  const int row = lane & 15;
  const int hi  = lane >> 4;             // 0: K pair {0,1}; 1: K pair {2,3}

  const float* __restrict__ aptr = A + (size_t)row * K + 2 * hi;
  const float* __restrict__ bptr = W + (size_t)(2 * hi) * N + col0 + row;

  v8f acc = {};
#pragma unroll 8
  for (int k = 0; k < K; k += 4) {
    v2f a = *(const v2f*)(aptr + k);     // A[row][k+2hi], A[row][k+2hi+1]
    v2f b;
    b.x = bptr[(size_t)k * N];           // W[k+2hi  ][col0+row]
    b.y = bptr[(size_t)k * N + N];       // W[k+2hi+1][col0+row]
    acc = __builtin_amdgcn_wmma_f32_16x16x4_f32(
        /*neg_a=*/false, a, /*neg_b=*/false, b,
        /*c_mod=*/(short)0, acc, /*reuse_a=*/false, /*reuse_b=*/false);
  }

  const float bv = bias[col0 + row];
#pragma unroll
  for (int r = 0; r < 8; ++r) {
    const int m = r + 8 * hi;            // D VGPR r -> M=r (lo lanes), r+8 (hi)
    C[(size_t)m * N + col0 + row] = acc[r] + bv;
  }
}

// ---------------------------------------------------------------------------
// NeoX RoPE on the first 32 dims of q and k inside the qkv workspace buffer.
// One block per (b,h); threads 0..15 each handle one rotation pair.
// ---------------------------------------------------------------------------
__global__ void rope_kernel(float* __restrict__ qkv, const int* __restrict__ pos_ids) {
  const int bh = blockIdx.x;
  const int b = bh / NH_, h = bh % NH_;
  const int i = threadIdx.x;
  if (i >= HALFROT) return;

  // inv_freq = 10000^(-2i/32) = exp(-(2i/32) * ln(10000))
  const float inv_freq = __expf(-(2.0f * i / (float)ROT_) * 9.210340371976184f);
  const float ang = (float)pos_ids[b] * inv_freq;
  float s, c;
  __sincosf(ang, &s, &c);

  float* qh = qkv + (size_t)b * QKV_N + h * HD_;
  float* kh = qkv + (size_t)b * QKV_N + HIDDEN_ + h * HD_;

  float x1 = qh[i], x2 = qh[i + HALFROT];
  qh[i]           = x1 * c - x2 * s;
  qh[i + HALFROT] = x1 * s + x2 * c;

  x1 = kh[i]; x2 = kh[i + HALFROT];
  kh[i]           = x1 * c - x2 * s;
  kh[i + HALFROT] = x1 * s + x2 * c;
}

// ---------------------------------------------------------------------------
// Flash-decode attention: one block per (b,h), 8 waves stream the 2049
// positions with online softmax. Lane d holds q/v dims [4d, 4d+4): a K row is
// 512 B = 32 lanes x float4, fully coalesced. New-token k/v come from the
// RoPE'd qkv buffer (position S).
// ---------------------------------------------------------------------------
__global__ __launch_bounds__(256)
void attn_kernel(const float* __restrict__ qkv,
                 const float* __restrict__ k_cache,
                 const float* __restrict__ v_cache,
                 float* __restrict__ attn_out) {
  constexpr int WAVES = 8;
  __shared__ float q_s[HD_];
  __shared__ float m_s[WAVES], l_s[WAVES];
  __shared__ float acc_s[WAVES][HD_];

  const int bh = blockIdx.x;
  const int b = bh / NH_, h = bh % NH_;
  const int tid = threadIdx.x;
  const int wave = tid >> 5, lane = tid & 31;

  const float scale = 0.08838834764831845f;  // 128^-0.5, folded into q
  if (tid < HD_) q_s[tid] = qkv[(size_t)b * QKV_N + h * HD_ + tid] * scale;
  __syncthreads();

  const float4 qv = *(const float4*)(q_s + lane * 4);

  float m = -3.0e38f, l = 0.0f;
  float4 acc = {0.f, 0.f, 0.f, 0.f};

  const float* knew = qkv + (size_t)b * QKV_N + HIDDEN_ + h * HD_;
  const float* vnew = qkv + (size_t)b * QKV_N + 2 * HIDDEN_ + h * HD_;

  for (int s = wave; s < S_ + 1; s += WAVES) {
    const float* kp;
    const float* vp;
    if (s < S_) {
      const size_t off = (((size_t)b * S_ + s) * NH_ + h) * HD_;
      kp = k_cache + off;
      vp = v_cache + off;
    } else {
      kp = knew;
      vp = vnew;
    }
    const float4 kv = *(const float4*)(kp + lane * 4);
    float part = qv.x * kv.x + qv.y * kv.y + qv.z * kv.z + qv.w * kv.w;
#pragma unroll
    for (int o = 16; o >= 1; o >>= 1) part += __shfl_xor(part, o, 32);
    const float score = part;  // scale already in q

    const float nm = fmaxf(m, score);
    const float corr = __expf(m - nm);
    const float p = __expf(score - nm);
    l = l * corr + p;

    const float4 vv = *(const float4*)(vp + lane * 4);
    acc.x = acc.x * corr + p * vv.x;
    acc.y = acc.y * corr + p * vv.y;
    acc.z = acc.z * corr + p * vv.z;
    acc.w = acc.w * corr + p * vv.w;
    m = nm;
  }

  if (lane == 0) { m_s[wave] = m; l_s[wave] = l; }
  acc_s[wave][lane * 4 + 0] = acc.x;
  acc_s[wave][lane * 4 + 1] = acc.y;
  acc_s[wave][lane * 4 + 2] = acc.z;
  acc_s[wave][lane * 4 + 3] = acc.w;
  __syncthreads();

  if (tid < HD_) {
    float M = -3.0e38f;
#pragma unroll
    for (int w = 0; w < WAVES; ++w) M = fmaxf(M, m_s[w]);
    float L = 0.0f, sum = 0.0f;
#pragma unroll
    for (int w = 0; w < WAVES; ++w) {
      const float f = __expf(m_s[w] - M);
      L += l_s[w] * f;
      sum += acc_s[w][tid] * f;
    }
    attn_out[(size_t)b * HIDDEN_ + h * HD_ + tid] = sum / L;
  }
}

// ---------------------------------------------------------------------------
extern "C" void kernel_launch(void* const* d_in, const int* in_sizes, int n_in,
                              void* d_out, int out_size, void* d_ws, size_t ws_size,
                              hipStream_t stream) {
  const float* hidden = (const float*)d_in[0];
  const int*   pos    = (const int*)d_in[1];
  const float* kc     = (const float*)d_in[2];
  const float* vc     = (const float*)d_in[3];
  const float* Wqkv   = (const float*)d_in[4];
  const float* bqkv   = (const float*)d_in[5];
  const float* Wd     = (const float*)d_in[6];
  const float* bd     = (const float*)d_in[7];
  float* out = (float*)d_out;

  float* qkv  = (float*)d_ws;                       // 16 x 12288
  float* attn = qkv + (size_t)B_ * QKV_N;           // 16 x 4096

  // 1) QKV projection: 768 tiles of 16 cols, 8 waves/block -> 96 blocks
  gemm16_wmma<<<QKV_N / 16 / 8, 256, 0, stream>>>(hidden, Wqkv, bqkv, qkv,
                                                  HIDDEN_, QKV_N);
  // 2) RoPE on q,k (in workspace)
  rope_kernel<<<B_ * NH_, 32, 0, stream>>>(qkv, pos);
  // 3) Flash-decode attention over 2048 cached + 1 new position
  attn_kernel<<<B_ * NH_, 256, 0, stream>>>(qkv, kc, vc, attn);
  // 4) Output projection: 256 tiles -> 32 blocks
  gemm16_wmma<<<HIDDEN_ / 16 / 8, 256, 0, stream>>>(attn, Wd, bd, out,
                                                    HIDDEN_, HIDDEN_);
}